// FakeImg_26929444946676
// MI455X (gfx1250) — compile-verified
//
#include <hip/hip_runtime.h>
#include <hip/hip_bf16.h>

// Toroidal 32x32 patch gather from a 3x4096x4096 f32 image.
// out[((n*3+c)*32+ph)*32+pw] = img[c][(ys[n]+ph)&4095][(xs[n]+pw)&4095]
//
// Pure bandwidth problem (~12 MB traffic @ 23.3 TB/s, zero FLOPs -> WMMA
// inapplicable). CDNA5 feature that fits: async global->LDS copies.
//  - grid (3, 512): c = blockIdx.x, n = blockIdx.y (no integer division)
//  - one workgroup per (n,c) tile; 256 threads = 8 wave32s
//  - per-lane GLOBAL_LOAD_ASYNC_TO_LDS_B32 gathers (ASYNCcnt), 4 in flight
//    per lane; power-of-two toroidal wrap is a single AND per coordinate
//  - drain with s_wait_asynccnt 0 + workgroup barrier
//  - emit output as ds_load_b128 + fully coalesced global_store_b128

#define IMG_H 4096
#define IMG_W 4096
#define P_H 32
#define P_W 32
#define N_CH 3
#define N_PATCH 512                 // B*S = 256*2
#define TILE_ELEMS (P_H * P_W)      // 1024 floats = 4 KB
#define THREADS 256

__global__ __launch_bounds__(THREADS)
void patch_gather_async_kernel(const float* __restrict__ img,
                               const int*   __restrict__ ys,
                               const int*   __restrict__ xs,
                               float*       __restrict__ out)
{
    __shared__ float tile[TILE_ELEMS];

    const int tid = threadIdx.x;
    const int c   = blockIdx.x;            // 0..2   (channel)
    const int n   = blockIdx.y;            // 0..511 (patch)
    const int bid = n * N_CH + c;          // output tile index

    // Block-uniform patch origin -> scalar loads (KMcnt path).
    const int y0 = ys[n];
    const int x0 = xs[n];

    const float* chan = img + (size_t)c * (size_t)IMG_H * (size_t)IMG_W;

    // Phase 1: per-lane async gathers straight into LDS (no VGPR data path).
    // 4 outstanding per lane before the wait -> latency overlap via ASYNCcnt.
#pragma unroll
    for (int it = 0; it < TILE_ELEMS / THREADS; ++it) {
        const int e  = tid + it * THREADS;     // element within tile
        const int ph = e >> 5;                 // row  (P_W == 32)
        const int pw = e & (P_W - 1);          // col
        const int y  = (y0 + ph) & (IMG_H - 1);   // toroidal wrap, pow2 dims
        const int x  = (x0 + pw) & (IMG_W - 1);

        const unsigned long long gaddr =
            (unsigned long long)(uintptr_t)(chan + (size_t)y * IMG_W + (size_t)x);
        const unsigned lds_addr =
            (unsigned)(uintptr_t)&tile[e];     // low 32 bits == LDS byte offset

        asm volatile("global_load_async_to_lds_b32 %0, %1, off"
                     :
                     : "v"(lds_addr), "v"(gaddr)
                     : "memory");
    }

    // Phase 2: drain this wave's async copies, then sync the workgroup so
    // every lane may read LDS words written by other waves' async loads.
    asm volatile("s_wait_asynccnt 0x0" ::: "memory");
    __syncthreads();

    // Phase 3: 128-bit LDS reads + fully coalesced 128-bit global stores.
    const float4 v = reinterpret_cast<const float4*>(tile)[tid];
    reinterpret_cast<float4*>(out)[(size_t)bid * (TILE_ELEMS / 4) + tid] = v;
}

extern "C" void kernel_launch(void* const* d_in, const int* in_sizes, int n_in,
                              void* d_out, int out_size, void* d_ws, size_t ws_size,
                              hipStream_t stream) {
    (void)in_sizes; (void)n_in; (void)out_size; (void)d_ws; (void)ws_size;

    const float* img = (const float*)d_in[0];   // [1,3,4096,4096] f32
    const int*   ys  = (const int*)d_in[1];     // [512] i32
    const int*   xs  = (const int*)d_in[2];     // [512] i32
    float*       out = (float*)d_out;           // [256,6,32,32] f32

    dim3 grid(N_CH, N_PATCH);                   // 3 x 512 = 1536 tiles
    dim3 block(THREADS);
    patch_gather_async_kernel<<<grid, block, 0, stream>>>(img, ys, xs, out);
}